// STU_66288525246559
// MI455X (gfx1250) — compile-verified
//
#include <hip/hip_runtime.h>
#include <hip/hip_bf16.h>
#include <stdint.h>

// STU forward for MI455X (gfx1250, wave32).
// Pipeline (all fp32, WMMA f32 16x16x4 for every matrix op):
//   1) fir_kernel:      p_conv = -depthwise_15tap(u, ker)
//   2) gemm_wmma_lds:   u_proj  = u      @ M_inputs   (16384x1024x256)
//   3) gemm_wmma_lds:   p_proj  = p_conv @ M_inputs   (16384x1024x256)
//   4) gemm_wmma:       phi     = filters@ M_filters  (4096x24x256, K=24)
//   5) conv_wmma:       s = p_proj + 2*sum_{s even<=t} phi[s,r]*u_proj[t-s,r]
//   6) gemm_wmma_lds:   out = s @ W_out               (16384x256x1024)
//
// gemm_wmma_lds stages A/B panels into LDS with GLOBAL_LOAD_ASYNC_TO_LDS_B32
// (ASYNCcnt-tracked DMA, inline asm: no gfx1250 builtin exposed), then feeds
// v_wmma_f32_16x16x4_f32 from ds_load. 64x64 C tile per 16-wave workgroup.

#define STU_B 4
#define STU_L 4096
#define STU_D 1024
#define STU_K 24
#define STU_R 256
#define STU_TAPS 15

#define TILE_M 64
#define TILE_N 64
#define TILE_K 16
#define A_PAD 2   // LDS row pad (floats) to break bank conflicts

typedef __attribute__((ext_vector_type(2))) float v2f;
typedef __attribute__((ext_vector_type(8))) float v8f;

// Issue one async memory->LDS dword copy (GVS addressing: saddr + 32-bit voff).
__device__ __forceinline__ void async_b32(unsigned lds_off, unsigned voff,
                                          const float* base) {
  asm volatile("global_load_async_to_lds_b32 %0, %1, %2"
               :: "v"(lds_off), "v"(voff), "s"(base) : "memory");
}
__device__ __forceinline__ void wait_async0() {
  asm volatile("s_wait_asynccnt 0x0" ::: "memory");
}
// Low 32 bits of a flat pointer to LDS == wave-relative LDS byte offset.
__device__ __forceinline__ unsigned lds_off32(const void* p) {
  return (unsigned)(uintptr_t)p;
}

// -------------------------------------------------------------------------
// 15-tap depthwise FIR:  p_conv[b,t,d] = -sum_j ker[j,d] * u[b, t+14-j, d]
// -------------------------------------------------------------------------
__global__ __launch_bounds__(256) void fir_kernel(const float* __restrict__ u,
                                                  const float* __restrict__ ker,
                                                  float* __restrict__ pconv) {
  unsigned idx = blockIdx.x * 256u + threadIdx.x;  // B*L*D threads (16Mi)
  unsigned d = idx & (STU_D - 1);
  unsigned bt = idx >> 10;             // b*L + t
  unsigned t = bt & (STU_L - 1);
  const float* ub = u + (bt - t) * STU_D + d;  // row (b*L) base + d
  float acc = 0.0f;
#pragma unroll
  for (int j = 0; j < STU_TAPS; ++j) {
    unsigned ti = t + (STU_TAPS - 1) - j;
    unsigned tc = ti < STU_L ? ti : (STU_L - 1);      // clamped, branch-free
    float uv = ub[tc * STU_D];
    uv = (ti < STU_L) ? uv : 0.0f;                    // v_cndmask, no EXEC div
    acc += ker[j * STU_D + d] * uv;
  }
  pconv[idx] = -acc;
}

// -------------------------------------------------------------------------
// LDS-tiled fp32 WMMA GEMM: C = A[rows x kdim] @ Bm[kdim x ncols]
// rows % 64 == 0, kdim % 16 == 0, ncols % 64 == 0.
// 16 waves (512 thr) per block -> 64x64 C tile; wave (wm,wn) owns 16x16.
// A/B panels staged via async-to-LDS DMA each K-chunk of 16.
// -------------------------------------------------------------------------
__global__ __launch_bounds__(512) void gemm_wmma_lds(const float* __restrict__ A,
                                                     const float* __restrict__ Bm,
                                                     float* __restrict__ C,
                                                     int rows, int kdim,
                                                     int ncols) {
  __shared__ float As[TILE_M][TILE_K + A_PAD];  // 64 x 18
  __shared__ float Bs[TILE_K][TILE_N];          // 16 x 64

  const int tid = threadIdx.x;
  const int lane = tid & 31;
  const int wave = tid >> 5;       // 0..15
  const int l2 = lane & 15;
  const int hi = lane >> 4;
  const int wm = wave >> 2;        // 0..3 : M sub-tile
  const int wn = wave & 3;         // 0..3 : N sub-tile
  const unsigned row0 = blockIdx.x * TILE_M;
  const unsigned col0 = blockIdx.y * TILE_N;

  // Staging assignment: 1024 elements per panel, 2 per thread.
  const int ea0 = tid, ea1 = tid + 512;
  // A element e -> (row e>>4, k e&15)
  unsigned avoff0 = ((row0 + (ea0 >> 4)) * (unsigned)kdim + (ea0 & 15)) * 4u;
  unsigned avoff1 = ((row0 + (ea1 >> 4)) * (unsigned)kdim + (ea1 & 15)) * 4u;
  const unsigned alds0 = lds_off32(&As[ea0 >> 4][ea0 & 15]);
  const unsigned alds1 = lds_off32(&As[ea1 >> 4][ea1 & 15]);
  // B element e -> (k e>>6, col e&63)
  unsigned bvoff0 = ((unsigned)(ea0 >> 6) * ncols + col0 + (ea0 & 63)) * 4u;
  unsigned bvoff1 = ((unsigned)(ea1 >> 6) * ncols + col0 + (ea1 & 63)) * 4u;
  const unsigned blds0 = lds_off32(&Bs[ea0 >> 6][ea0 & 63]);
  const unsigned blds1 = lds_off32(&Bs[ea1 >> 6][ea1 & 63]);
  const unsigned bstep = (unsigned)TILE_K * ncols * 4u;

  v8f acc = {};
  for (int kk = 0; kk < kdim; kk += TILE_K) {
    async_b32(alds0, avoff0, A);
    async_b32(alds1, avoff1, A);
    async_b32(blds0, bvoff0, Bm);
    async_b32(blds1, bvoff1, Bm);
    avoff0 += TILE_K * 4u;
    avoff1 += TILE_K * 4u;
    bvoff0 += bstep;
    bvoff1 += bstep;
    wait_async0();        // this wave's DMA done
    __syncthreads();      // everyone's DMA visible
#pragma unroll
    for (int kc = 0; kc < 4; ++kc) {
      // A 16x4 frag: lane (l2,hi), VGPR j -> K = 4*kc + 2*hi + j (b64 ds load)
      v2f a = *(const v2f*)&As[16 * wm + l2][4 * kc + 2 * hi];
      // B 4x16 frag: lane -> col, VGPR j -> row 4*kc + 2*hi + j
      v2f b;
      b.x = Bs[4 * kc + 2 * hi][16 * wn + l2];
      b.y = Bs[4 * kc + 2 * hi + 1][16 * wn + l2];
      acc = __builtin_amdgcn_wmma_f32_16x16x4_f32(false, a, false, b, (short)0,
                                                  acc, false, false);
    }
    __syncthreads();      // reads done before next-chunk DMA overwrites
  }
  // C 16x16: VGPR v, lane -> (M = v + 8*hi, N = l2)
  float* crow =
      C + (size_t)(row0 + 16 * wm + 8 * hi) * ncols + col0 + 16 * wn + l2;
#pragma unroll
  for (int v = 0; v < 8; ++v) crow[(size_t)v * ncols] = acc[v];
}

// -------------------------------------------------------------------------
// Simple fp32 WMMA GEMM (no LDS) for small-K shapes: kdim % 4 == 0.
// 4 waves per block; each wave owns a 16x16 C tile.
// -------------------------------------------------------------------------
__global__ __launch_bounds__(128) void gemm_wmma(const float* __restrict__ A,
                                                 const float* __restrict__ Bm,
                                                 float* __restrict__ C,
                                                 int rows, int kdim, int ncols) {
  const int wave = threadIdx.x >> 5;
  const int lane = threadIdx.x & 31;
  const int l2 = lane & 15;
  const int hi = lane >> 4;
  const unsigned row0 = blockIdx.x * 16;
  const unsigned col0 = blockIdx.y * 64 + wave * 16;

  const float* arow = A + (size_t)(row0 + l2) * kdim;
  const float* bcol = Bm + (col0 + l2);
  v8f c = {};
  for (int kk = 0; kk < kdim; kk += 4) {
    v2f a = *(const v2f*)(arow + kk + 2 * hi);
    v2f b;
    b.x = bcol[(unsigned)(kk + 2 * hi) * ncols];
    b.y = bcol[(unsigned)(kk + 2 * hi + 1) * ncols];
    c = __builtin_amdgcn_wmma_f32_16x16x4_f32(false, a, false, b, (short)0, c,
                                              false, false);
  }
  float* crow = C + (size_t)(row0 + 8 * hi) * ncols + col0 + l2;
#pragma unroll
  for (int v = 0; v < 8; ++v) crow[(unsigned)v * ncols] = c[v];
}

// -------------------------------------------------------------------------
// Spectral conv as per-(b,r,parity) block-Toeplitz WMMA (parity-folded):
//   x[q] = u_proj[b, 2q+p, r],  h[m] = 2*phi[2m, r]
//   y_i  = sum_k H_k x_{i-k},   H_k[a,c] = h[16k + a - c]  (0 if m<0)
// One wave: 16 output tiles (C cols) x 16 rows; store fused with +p_proj.
// grid = (8 tile-groups, R, B*2), block = 32 (EXEC all ones for WMMA).
// -------------------------------------------------------------------------
__global__ __launch_bounds__(32) void conv_wmma(const float* __restrict__ uproj,
                                                const float* __restrict__ phi,
                                                const float* __restrict__ pproj,
                                                float* __restrict__ sbuf) {
  const int lane = threadIdx.x & 31;
  const int l2 = lane & 15;
  const int hi = lane >> 4;
  const int g = blockIdx.x;        // output tile group: tiles 16g .. 16g+15
  const int r = blockIdx.y;        // channel
  const int p = blockIdx.z & 1;    // parity
  const int b = blockIdx.z >> 1;   // batch

  const float* hb = phi + r;                                   // h[m]=2*hb[2m*R]
  const float* xb = uproj + ((unsigned)b * STU_L + p) * STU_R + r;  // x[q]=xb[2q*R]

  const int i = 16 * g + l2;       // this lane's output tile (C column)
  v8f c = {};
  const int kmax = 16 * g + 15;
  for (int k = 0; k <= kmax; ++k) {
    const int src = i - k;                 // input tile for this column
    const int srcc = src > 0 ? src : 0;    // clamped (branch-free)
    const float xm = (src >= 0) ? 1.0f : 0.0f;
#pragma unroll
    for (int kc = 0; kc < 4; ++kc) {
      const int c0 = 4 * kc + 2 * hi;
      const int m0 = 16 * k + l2 - c0;     // <= 2047 always
      const int m1 = m0 - 1;
      const unsigned m0c = (unsigned)(m0 > 0 ? m0 : 0);
      const unsigned m1c = (unsigned)(m1 > 0 ? m1 : 0);
      v2f a;
      a.x = (m0 >= 0) ? 2.0f * hb[2u * m0c * STU_R] : 0.0f;
      a.y = (m1 >= 0) ? 2.0f * hb[2u * m1c * STU_R] : 0.0f;
      const unsigned q0 = 16u * (unsigned)srcc + (unsigned)c0;
      v2f bv;
      bv.x = xm * xb[2u * q0 * STU_R];
      bv.y = xm * xb[(2u * q0 + 2u) * STU_R];
      c = __builtin_amdgcn_wmma_f32_16x16x4_f32(false, a, false, bv, (short)0,
                                                c, false, false);
    }
  }
#pragma unroll
  for (int v = 0; v < 8; ++v) {
    const unsigned tau = 16u * (unsigned)i + (unsigned)(v + 8 * hi);
    const unsigned off = ((unsigned)b * STU_L + 2u * tau + (unsigned)p) * STU_R
                         + (unsigned)r;
    sbuf[off] = pproj[off] + c[v];
  }
}

// -------------------------------------------------------------------------
extern "C" void kernel_launch(void* const* d_in, const int* in_sizes, int n_in,
                              void* d_out, int out_size, void* d_ws,
                              size_t ws_size, hipStream_t stream) {
  const float* u       = (const float*)d_in[0];  // (B,L,D)
  const float* filters = (const float*)d_in[1];  // (L,K)
  const float* ker     = (const float*)d_in[2];  // (15,D)
  const float* Min     = (const float*)d_in[3];  // (D,R)
  const float* Mfil    = (const float*)d_in[4];  // (K,R)
  const float* Wout    = (const float*)d_in[5];  // (R,D)
  float* out = (float*)d_out;                    // (B,L,D)

  char* ws = (char*)d_ws;
  const size_t BLD = (size_t)STU_B * STU_L * STU_D;  // 16 Mi elems, 64 MB
  const size_t BLR = (size_t)STU_B * STU_L * STU_R;  //  4 Mi elems, 16 MB
  float* pconv   = (float*)ws;                               // 64 MB
  float* uproj   = (float*)(ws + BLD * 4);                   // 16 MB
  float* pproj   = (float*)(ws + BLD * 4 + BLR * 4);         // 16 MB
  float* phiproj = (float*)(ws + BLD * 4 + 2 * BLR * 4);     //  4 MB
  float* sbuf    = pconv;  // pconv dead after step 3; reuse for conv output

  const int rowsBL = STU_B * STU_L;

  // 1) depthwise FIR
  fir_kernel<<<dim3((unsigned)(BLD / 256)), dim3(256), 0, stream>>>(u, ker,
                                                                    pconv);
  // 2,3) projections: (B*L x D) @ (D x R), LDS-tiled async WMMA GEMM
  gemm_wmma_lds<<<dim3(rowsBL / TILE_M, STU_R / TILE_N), dim3(512), 0, stream>>>(
      u, Min, uproj, rowsBL, STU_D, STU_R);
  gemm_wmma_lds<<<dim3(rowsBL / TILE_M, STU_R / TILE_N), dim3(512), 0, stream>>>(
      pconv, Min, pproj, rowsBL, STU_D, STU_R);
  // 4) filter projection: (L x 24) @ (24 x R), small-K direct WMMA GEMM
  gemm_wmma<<<dim3(STU_L / 16, STU_R / 64), dim3(128), 0, stream>>>(
      filters, Mfil, phiproj, STU_L, STU_K, STU_R);
  // 5) block-Toeplitz spectral conv (+ p_proj), parity-folded
  conv_wmma<<<dim3(8, STU_R, STU_B * 2), dim3(32), 0, stream>>>(
      uproj, phiproj, pproj, sbuf);
  // 6) output projection: (B*L x R) @ (R x D)
  gemm_wmma_lds<<<dim3(rowsBL / TILE_M, STU_D / TILE_N), dim3(512), 0, stream>>>(
      sbuf, Wout, out, rowsBL, STU_R, STU_D);
}